// APPNPModel_16492674417393
// MI455X (gfx1250) — compile-verified
//
#include <hip/hip_runtime.h>

typedef __attribute__((ext_vector_type(2))) float v2f;
typedef __attribute__((ext_vector_type(8))) float v8f;

#define N_NODES  50000
#define N_FEATS  2048
#define NNZ_CNT  1000000
#define N_EDGE   800000
#define HID      256
#define LABELS   64
#define ALPHA_F  0.1f
#define N_ITERS  10

// ---------------------------------------------------------------- zero fill
__global__ void k_zero4(float4* __restrict__ p, unsigned n4) {
    unsigned i = blockIdx.x * blockDim.x + threadIdx.x;
    if (i < n4) p[i] = make_float4(0.f, 0.f, 0.f, 0.f);
}

// --------------------------------------------- feature SpMM: h1 += S @ W1
// one thread per (nnz, 4-col group): 64 threads cover the 256 hidden cols
__global__ void k_spmm_feat(const int* __restrict__ frow,
                            const int* __restrict__ fcol,
                            const float* __restrict__ fval,
                            const float* __restrict__ W1,
                            float* __restrict__ h1) {
    unsigned gid = blockIdx.x * blockDim.x + threadIdx.x;
    unsigned e = gid >> 6;
    if (e >= NNZ_CNT) return;
    unsigned cg = (gid & 63u) << 2;           // 0,4,...,252
    int r = frow[e];
    int c = fcol[e];
    float v = fval[e];
    const float4 w = *(const float4*)(W1 + (size_t)c * HID + cg);
    float* dst = h1 + (size_t)r * HID + cg;
    atomicAdd(dst + 0, v * w.x);
    atomicAdd(dst + 1, v * w.y);
    atomicAdd(dst + 2, v * w.z);
    atomicAdd(dst + 3, v * w.w);
}

// ------------------------- dense layer: h2 = relu(h1 + b1) @ W2 + b2 (WMMA)
// 256 threads = 8 waves; wave computes one 16x16 fp32 tile via
// V_WMMA_F32_16X16X4_F32 (full fp32 precision on the matrix pipe).
// Block covers rows [32*bx, 32*bx+32), all 64 label columns.
// N_NODES % 16 == 0, tiles start at multiples of 16 -> validity is
// wave-uniform; OOB waves exit before any WMMA (EXEC stays all-1s for
// every wave that executes WMMA, per ISA 7.12 restrictions).
__global__ void k_gemm_wmma(const float* __restrict__ h1,
                            const float* __restrict__ b1,
                            const float* __restrict__ W2,
                            const float* __restrict__ b2,
                            float* __restrict__ h2) {
    const int wave = threadIdx.x >> 5;
    const int lane = threadIdx.x & 31;
    const int m0   = blockIdx.x * 32 + ((wave >> 2) << 4); // +0 or +16
    if (m0 >= N_NODES) return;                 // wave-uniform tile guard
    const int n0   = (wave & 3) << 4;          // 0,16,32,48
    const int half = lane >> 4;                // 0 or 1 (K pairing)
    const int l15  = lane & 15;                // A: row in tile / B,D: col

    const float* h1row = h1 + (size_t)(m0 + l15) * HID;
    const float* w2col = W2 + n0 + l15;

    v8f acc = {};
#pragma unroll 4
    for (int kk = 0; kk < HID; kk += 4) {
        const int k = kk + (half << 1);
        // A fragment: relu(h1 + b1) for K = k, k+1 (single b64 loads)
        const v2f hv = *(const v2f*)(h1row + k);
        const v2f bv = *(const v2f*)(b1 + k);
        v2f a;
        a.x = fmaxf(hv.x + bv.x, 0.f);
        a.y = fmaxf(hv.y + bv.y, 0.f);
        // B fragment: W2[k][n0+l15], W2 row-major stride LABELS
        v2f b;
        b.x = w2col[(size_t)k * LABELS];
        b.y = w2col[(size_t)(k + 1) * LABELS];
        acc = __builtin_amdgcn_wmma_f32_16x16x4_f32(
            /*neg_a=*/false, a, /*neg_b=*/false, b,
            /*c_mod=*/(short)0, acc, /*reuse_a=*/false, /*reuse_b=*/false);
    }

    // D layout: acc[r] = D[8*half + r][l15]
    const int   mbase = m0 + (half << 3);
    const float bias  = b2[n0 + l15];
    float* orow = h2 + (size_t)mbase * LABELS + n0 + l15;
#pragma unroll
    for (int r = 0; r < 8; ++r)
        orow[(size_t)r * LABELS] = acc[r] + bias;
}

// ----------------------------------------------------- zn = ALPHA * h2
__global__ void k_scale4(const float4* __restrict__ src,
                         float4* __restrict__ dst, unsigned n4) {
    unsigned i = blockIdx.x * blockDim.x + threadIdx.x;
    if (i < n4) {
        float4 v = src[i];
        dst[i] = make_float4(ALPHA_F * v.x, ALPHA_F * v.y,
                             ALPHA_F * v.z, ALPHA_F * v.w);
    }
}

// --------------------- edge scatter: zdst[erow] += (1-a)*w * zsrc[ecol]
// 16 threads per edge, 4 cols (float4 gather) per thread
__global__ void k_edge_scatter(const int* __restrict__ erow,
                               const int* __restrict__ ecol,
                               const float* __restrict__ ew,
                               const float* __restrict__ zsrc,
                               float* __restrict__ zdst) {
    unsigned gid = blockIdx.x * blockDim.x + threadIdx.x;
    unsigned e = gid >> 4;
    if (e >= N_EDGE) return;
    unsigned c = (gid & 15u) << 2;           // 0,4,...,60
    int rd = erow[e];
    int rs = ecol[e];
    float w = (1.0f - ALPHA_F) * ew[e];
    const float4 z = *(const float4*)(zsrc + (size_t)rs * LABELS + c);
    float* dst = zdst + (size_t)rd * LABELS + c;
    atomicAdd(dst + 0, w * z.x);
    atomicAdd(dst + 1, w * z.y);
    atomicAdd(dst + 2, w * z.z);
    atomicAdd(dst + 3, w * z.w);
}

// ------------------------------------------- log_softmax, one wave32 / row
__global__ void k_log_softmax(const float* __restrict__ z,
                              float* __restrict__ out) {
    const int row  = blockIdx.x * 8 + (threadIdx.x >> 5);
    const int lane = threadIdx.x & 31;
    if (row >= N_NODES) return;
    const float* zr = z + (size_t)row * LABELS;
    float a = zr[lane];
    float b = zr[lane + 32];
    float m = fmaxf(a, b);
#pragma unroll
    for (int off = 16; off > 0; off >>= 1)
        m = fmaxf(m, __shfl_xor(m, off, 32));
    float s = __expf(a - m) + __expf(b - m);
#pragma unroll
    for (int off = 16; off > 0; off >>= 1)
        s += __shfl_xor(s, off, 32);
    const float ls = __logf(s);
    float* orow = out + (size_t)row * LABELS;
    orow[lane]      = a - m - ls;
    orow[lane + 32] = b - m - ls;
}

// ---------------------------------------------------------------- launcher
extern "C" void kernel_launch(void* const* d_in, const int* in_sizes, int n_in,
                              void* d_out, int out_size, void* d_ws, size_t ws_size,
                              hipStream_t stream) {
    const int*   frow = (const int*)d_in[0];
    const int*   fcol = frow + NNZ_CNT;
    const float* fval = (const float*)d_in[1];
    const int*   erow = (const int*)d_in[2];
    const int*   ecol = erow + N_EDGE;
    const float* ew   = (const float*)d_in[3];
    const float* W1   = (const float*)d_in[4];
    const float* b1   = (const float*)d_in[5];
    const float* W2   = (const float*)d_in[6];
    const float* b2   = (const float*)d_in[7];
    float* out = (float*)d_out;

    float* h1 = (float*)d_ws;                           // 50000*256 f32 (51.2 MB)
    float* h2 = h1 + (size_t)N_NODES * HID;             // 50000*64
    float* zA = h2 + (size_t)N_NODES * LABELS;          // 50000*64
    float* zB = zA + (size_t)N_NODES * LABELS;          // 50000*64

    const unsigned h1_q = (unsigned)((size_t)N_NODES * HID / 4);     // 3.2M
    const unsigned z_q  = (unsigned)((size_t)N_NODES * LABELS / 4);  // 0.8M

    // h1 = 0
    k_zero4<<<(h1_q + 255) / 256, 256, 0, stream>>>((float4*)h1, h1_q);
    // h1 += S_feat @ W1
    {
        unsigned nthr = (unsigned)NNZ_CNT * 64u;
        k_spmm_feat<<<(nthr + 255) / 256, 256, 0, stream>>>(frow, fcol, fval, W1, h1);
    }
    // h2 = relu(h1 + b1) @ W2 + b2   (WMMA fp32)
    k_gemm_wmma<<<(N_NODES + 31) / 32, 256, 0, stream>>>(h1, b1, W2, b2, h2);

    // APPNP: z <- (1-a) A z + a h2, 10 iters, ping-pong zA/zB
    const float* src = h2;
    for (int it = 0; it < N_ITERS; ++it) {
        float* dst = (it & 1) ? zB : zA;
        k_scale4<<<(z_q + 255) / 256, 256, 0, stream>>>((const float4*)h2,
                                                        (float4*)dst, z_q);
        {
            unsigned nthr = (unsigned)N_EDGE * 16u;
            k_edge_scatter<<<(nthr + 255) / 256, 256, 0, stream>>>(
                erow, ecol, ew, src, dst);
        }
        src = dst;
    }

    // out = log_softmax(z)
    k_log_softmax<<<N_NODES / 8, 256, 0, stream>>>(src, out);
}